// ChannelAttention_50998441672727
// MI455X (gfx1250) — compile-verified
//
#include <hip/hip_runtime.h>
#include <math.h>

// ChannelAttention on gfx1250: fp32 WMMA (V_WMMA_F32_16X16X4_F32) path.
#define FQ  128
#define TT  2048
#define NH  8
#define DKC 32
#define DVC 32
#define CH  128      // EMA chunk length per wave
#define WU  192      // EMA warm-up (0.9^192 ~ 1.6e-9, below fp32 noise)
#define MOM 0.9f
#define OMM (1.0f - MOM)
#define EPSV 1e-8f

typedef __attribute__((ext_vector_type(2))) float v2f;
typedef __attribute__((ext_vector_type(4))) float v4f;
typedef __attribute__((ext_vector_type(8))) float v8f;

__device__ __forceinline__ v8f wmma_f32_16x16x4(v2f a, v2f b, v8f c) {
  return __builtin_amdgcn_wmma_f32_16x16x4_f32(false, a, false, b,
                                               (short)0, c, false, false);
}

// Kernel 1: per (f, t-chunk) wave. attT[l][q] = Kd * Q^T via 8 chained fp32
// WMMAs. Branch-free loads (columns 8-15 duplicate 0-7; t==0 repair on D),
// EMA scan in registers, ds_swizzle cross-half reduction for sum-of-squares,
// LDS repack for coalesced b128 stores of the att intermediate.
__global__ __launch_bounds__(256)
void ca_att_ema(const float* __restrict__ Kp, const float* __restrict__ Qp,
                float* __restrict__ att, float* __restrict__ psum) {
  __shared__ __align__(16) float lds[8 * 128];     // 128 floats per wave
  const int tid  = threadIdx.x;
  const int wv   = tid >> 5;
  const int lane = tid & 31;
  const int bpf  = (TT / CH) / 8;                  // blocks per f
  const int f    = blockIdx.x / bpf;
  const int cb   = blockIdx.x % bpf;
  const int chunk = cb * 8 + wv;                   // 0 .. TT/CH-1
  const int g    = lane >> 4;
  const int n    = lane & 15;                      // A row (=l) / B col (=q)
  const int c0   = g * 2;                          // K sub-offset per half
  const int tBeg = chunk * CH;
  const int tSt  = (chunk == 0) ? 0 : (tBeg - WU);
  float* lw = lds + wv * 128;
  const size_t fbase = (size_t)f * TT;

  const int arow = (n < 8) ? n : (n - 8);          // Kd row -> k row
  const int adel = (n < 8) ? 0 : 1;                // from timestep t - adel
  const int brow = n & 7;                          // cols 8-15 duplicate 0-7

  v8f state = {};                                  // EMA state, WMMA D layout

  for (int t = tSt; t < tBeg + CH; ++t) {
    const int ta = (t >= adel) ? (t - adel) : 0;   // clamped (repaired below)
    const float* ar = Kp + ((fbase + ta) * NH + arow) * DKC + c0;
    const float* br = Qp + ((fbase + t) * NH + brow) * DKC + c0;
    __builtin_prefetch(ar + 16 * NH * DKC, 0, 1);  // global_prefetch_b8
    __builtin_prefetch(br + 16 * NH * DKC, 0, 1);

    v2f a[8], b[8];                                // batch all 16 b64 loads
#pragma unroll
    for (int kc = 0; kc < 8; ++kc) {
      a[kc] = *(const v2f*)(ar + kc * 4);
      b[kc] = *(const v2f*)(br + kc * 4);
    }
    v8f acc = {};
#pragma unroll
    for (int kc = 0; kc < 8; ++kc) acc = wmma_f32_16x16x4(a[kc], b[kc], acc);

    if (t == 0) {                // Kd rows 8-15 invalid at t==0 -> zero D rows
      if (lane >= 16) {          // lanes 16-31 hold D rows 8-15
#pragma unroll
        for (int j = 0; j < 8; ++j) acc[j] = 0.0f;
      }
    }

#pragma unroll
    for (int j = 0; j < 8; ++j) state[j] = MOM * state[j] + OMM * acc[j];

    if (t >= tBeg) {
      float ss = 0.0f;
#pragma unroll
      for (int j = 0; j < 8; ++j) ss += state[j] * state[j];
      ss += __int_as_float(__builtin_amdgcn_ds_swizzle(
                __float_as_int(ss), 0x401F));      // SWAPX16
      if (lane < 8) psum[(fbase + t) * NH + lane] = ss;

      if (n < 8) {                                 // repack attT -> [l*8+q]
#pragma unroll
        for (int j = 0; j < 8; ++j) lw[(j + 8 * g) * 8 + n] = state[j];
      }
      v4f pk = *(const v4f*)(lw + lane * 4);       // same-wave LDS in-order
      *(v4f*)(att + (fbase + t) * 128 + lane * 4) = pk;
    }
  }
}

// Kernel 2: norm[t,q] = sqrt(sum_f psum[f,t,q]); coalesced per-f reads.
__global__ __launch_bounds__(256)
void ca_norm(const float* __restrict__ psum, float* __restrict__ nrm) {
  const int i = blockIdx.x * 256 + threadIdx.x;
  if (i >= TT * NH) return;
  float s = 0.0f;
#pragma unroll 4
  for (int f = 0; f < FQ; ++f) s += psum[(size_t)f * (TT * NH) + i];
  nrm[i] = sqrtf(s);
}

// Kernel 3: out = (att/(norm+eps)) * Vd via fp32 WMMA, two d-halves.
__global__ __launch_bounds__(256)
void ca_out(const float* __restrict__ Vp, const float* __restrict__ att,
            const float* __restrict__ nrm, float* __restrict__ outp) {
  const int tid  = threadIdx.x;
  const int lane = tid & 31;
  const int gw   = blockIdx.x * 8 + (tid >> 5);
  const int f    = gw >> 7;                        // 128 waves per f
  const int tb   = gw & 127;                       // 16 timesteps per wave
  const int g    = lane >> 4;
  const int n    = lane & 15;
  const int c0   = g * 2;
  const int qa   = n & 7;                          // A rows 8-15 duplicate 0-7
  const size_t fbase = (size_t)f * TT;

  for (int t = tb * 16; t < tb * 16 + 16; ++t) {
    const float* arow = att + (fbase + t) * 128;   // layout l*8+q
    const int tm = (t > 0) ? (t - 1) : 0;

    float sc[8];                                   // wave-uniform -> s_load
#pragma unroll
    for (int j = 0; j < 8; ++j) sc[j] = 1.0f / (nrm[t * NH + j] + EPSV);

    v2f a[4], bA[4], bB[4];
#pragma unroll
    for (int lc = 0; lc < 4; ++lc) {
      const int la = lc * 4 + c0;                  // l chunk rows (la, la+1)
      const int lb = la + 1;
      a[lc].x = arow[la * 8 + qa];
      a[lc].y = arow[lb * 8 + qa];
      // lc<2 -> Vd rows from v[t]; lc>=2 -> from v[t-1] (compile-time select)
      const int   ts = (lc < 2) ? t : tm;
      const int   r0 = (lc < 2) ? la : la - 8;
      const int   r1 = (lc < 2) ? lb : lb - 8;
      const float* v0 = Vp + ((fbase + ts) * NH + r0) * DVC;
      const float* v1 = Vp + ((fbase + ts) * NH + r1) * DVC;
      bA[lc].x = v0[n];      bA[lc].y = v1[n];
      bB[lc].x = v0[16 + n]; bB[lc].y = v1[16 + n];
      if (lc >= 2 && t == 0) {                     // Vd rows 8-15 zero at t==0
        bA[lc].x = 0.0f; bA[lc].y = 0.0f;
        bB[lc].x = 0.0f; bB[lc].y = 0.0f;
      }
    }

    v8f acc0 = {};
    v8f acc1 = {};
#pragma unroll
    for (int lc = 0; lc < 4; ++lc) {
      acc0 = wmma_f32_16x16x4(a[lc], bA[lc], acc0);   // d = 0..15
      acc1 = wmma_f32_16x16x4(a[lc], bB[lc], acc1);   // d = 16..31
    }

    if (g == 0) {                                  // lanes 0-15 hold rows 0-7
#pragma unroll
      for (int j = 0; j < 8; ++j) {
        float* orow = outp + ((fbase + t) * NH + j) * DVC;
        orow[n]      = acc0[j] * sc[j];
        orow[16 + n] = acc1[j] * sc[j];
      }
    }
  }
}

extern "C" void kernel_launch(void* const* d_in, const int* in_sizes, int n_in,
                              void* d_out, int out_size, void* d_ws, size_t ws_size,
                              hipStream_t stream) {
  (void)in_sizes; (void)n_in; (void)out_size; (void)ws_size;
  const float* k = (const float*)d_in[0];
  const float* q = (const float*)d_in[1];
  const float* v = (const float*)d_in[2];
  float* out = (float*)d_out;

  char* w = (char*)d_ws;
  float* att  = (float*)w;                                            // 128 MB
  float* psum = (float*)(w + (size_t)FQ * TT * 128 * sizeof(float));  // 8 MB
  float* nrm  = (float*)(w + (size_t)FQ * TT * 128 * sizeof(float)
                           + (size_t)FQ * TT * NH * sizeof(float));   // 64 KB

  ca_att_ema<<<FQ * ((TT / CH) / 8), 256, 0, stream>>>(k, q, att, psum); // 2048 waves
  ca_norm<<<(TT * NH + 255) / 256, 256, 0, stream>>>(psum, nrm);
  ca_out<<<(FQ * (TT / 16)) / 8, 256, 0, stream>>>(v, att, nrm, out);
}